// MaskedCrossAttention_26938034881161
// MI455X (gfx1250) — compile-verified
//
#include <hip/hip_runtime.h>
#include <cfloat>

#define HEADS 20
#define DHEAD 64
#define NQ    4096
#define BSZ   8
#define QD    1280
#define CDIM  1024
#define INNER 1280   // HEADS*DHEAD
#define CTX   85

typedef __attribute__((ext_vector_type(2))) float v2f;
typedef __attribute__((ext_vector_type(8))) float v8f;

// D = A(16x4) * B(4x16) + C, fp32 WMMA (wave32).
__device__ inline v8f wmma4(v2f a, v2f b, v8f c) {
  return __builtin_amdgcn_wmma_f32_16x16x4_f32(
      /*neg_a=*/false, a, /*neg_b=*/false, b,
      /*c_mod=*/(short)0, c, /*reuse_a=*/false, /*reuse_b=*/false);
}

// -------------------------------------------------------------------------
// fg map: bicubic 256->64 is a fixed 4-tap stencil (t == 0.5 everywhere).
// fg[b*4096 + o*64 + p] = (resized != 0)
// -------------------------------------------------------------------------
__global__ __launch_bounds__(256) void fg_kernel(const int* __restrict__ mask,
                                                 int* __restrict__ fg) {
  int tid = blockIdx.x * blockDim.x + threadIdx.x;
  if (tid >= BSZ * 64 * 64) return;
  int b = tid / 4096;
  int o = (tid % 4096) / 64;
  int p = tid % 64;
  const float c[4] = {-0.09375f, 0.59375f, 0.59375f, -0.09375f};
  const int* mb = mask + (size_t)b * 256 * 256;
  float r = 0.f;
  #pragma unroll
  for (int dw = 0; dw < 4; ++dw) {
    float t = 0.f;
    #pragma unroll
    for (int dh = 0; dh < 4; ++dh)
      t += c[dh] * (float)mb[(4 * o + dh) * 256 + (4 * p + dw)];
    r += t * c[dw];
  }
  fg[tid] = (r != 0.0f) ? 1 : 0;
}

// -------------------------------------------------------------------------
// fp32 WMMA GEMM: C[M,N] = A[M,K] * B[K,N] (+ bias[N]).
// Block = 256 threads = 8 waves arranged 2(M) x 4(N); block tile 128 x 256.
// Wave tile 64(M) x 64(N) = 16 accumulators; K stepped by 4.
// Inner loop is branch-free: OOB A rows are clamped to M-1 (their results
// are never stored). N % 256 == 0, K % 4 == 0 required.
// -------------------------------------------------------------------------
__global__ __launch_bounds__(256) void gemm_kernel(
    const float* __restrict__ A, const float* __restrict__ Bm,
    const float* __restrict__ bias, float* __restrict__ C,
    int M, int K, int N) {
  int wv    = threadIdx.x >> 5;
  int lane  = threadIdx.x & 31;
  int l15   = lane & 15;
  int half  = lane >> 4;
  int khalf = half << 1;               // this half-wave's K sub-offset
  int wm = wv >> 2, wn = wv & 3;
  int m0 = blockIdx.y * 128 + wm * 64;
  int n0 = blockIdx.x * 256 + wn * 64;

  // A row pointers for the 4 M-subtiles, clamped in-range (branch-free loop).
  const float* aptr[4];
  #pragma unroll
  for (int ms = 0; ms < 4; ++ms) {
    int arow = m0 + ms * 16 + l15;
    aptr[ms] = A + (size_t)(arow < M ? arow : M - 1) * K;
  }

  v8f acc[4][4] = {};
  for (int k = 0; k < K; k += 4) {
    v2f a[4];
    #pragma unroll
    for (int ms = 0; ms < 4; ++ms)
      a[ms] = *(const v2f*)(aptr[ms] + k + khalf);     // global_load_b64
    const float* b0 = Bm + (size_t)(k + khalf) * N + n0 + l15;
    #pragma unroll
    for (int nt = 0; nt < 4; ++nt) {
      v2f bf;
      bf.x = b0[nt * 16];
      bf.y = b0[nt * 16 + N];
      #pragma unroll
      for (int ms = 0; ms < 4; ++ms)
        acc[ms][nt] = wmma4(a[ms], bf, acc[ms][nt]);
    }
  }

  #pragma unroll
  for (int ms = 0; ms < 4; ++ms) {
    #pragma unroll
    for (int nt = 0; nt < 4; ++nt) {
      #pragma unroll
      for (int v = 0; v < 8; ++v) {
        int row = m0 + ms * 16 + v + 8 * half;
        if (row < M) {
          int col = n0 + nt * 16 + l15;
          float r = acc[ms][nt][v];
          if (bias) r += bias[col];
          C[(size_t)row * N + col] = r;
        }
      }
    }
  }
}

// -------------------------------------------------------------------------
// Fused masked attention: one wave per (b, h, 16-query tile).
// sim = (Q Kt)*0.125 with mask, softmax over 85 cols, out = P V.
// All global loads are clamped (branch-free): clamped K columns get masked
// to -inf, clamped V rows multiply P == 0.
// -------------------------------------------------------------------------
__global__ __launch_bounds__(256) void attn_kernel(
    const float* __restrict__ q, const float* __restrict__ kbuf,
    const float* __restrict__ vbuf, const int* __restrict__ fg,
    float* __restrict__ out) {
  __shared__ float plds[8][16 * 96];   // per-wave P tile, 96-padded K dim

  int wv    = threadIdx.x >> 5;
  int lane  = threadIdx.x & 31;
  int l15   = lane & 15;
  int half  = lane >> 4;
  int khalf = half << 1;

  int wid = blockIdx.x * 8 + wv;
  int b   = wid / (HEADS * (NQ / 16));
  int rem = wid % (HEADS * (NQ / 16));
  int h   = rem / (NQ / 16);
  int i0  = (rem % (NQ / 16)) * 16;

  // ---- sim = Q * K^T : 6 j-tiles of 16, K-dim = 64 stepped by 4 ----
  const float* qbase = q + ((size_t)(b * NQ + i0 + l15)) * INNER + h * DHEAD;
  const float* kptr[6];
  #pragma unroll
  for (int jt = 0; jt < 6; ++jt) {
    int j = jt * 16 + l15;
    int jc = j < CTX ? j : CTX - 1;    // clamp; masked to -inf later
    kptr[jt] = kbuf + ((size_t)(b * CTX + jc)) * INNER + h * DHEAD;
  }

  v8f s[6] = {};
  for (int kb = 0; kb < DHEAD; kb += 4) {
    v2f a = *(const v2f*)(qbase + kb + khalf);          // global_load_b64
    #pragma unroll
    for (int jt = 0; jt < 6; ++jt) {
      v2f bf = *(const v2f*)(kptr[jt] + kb + khalf);    // global_load_b64
      s[jt] = wmma4(a, bf, s[jt]);
    }
  }

  // ---- scale + custom mask ----
  int fgv[8];
  #pragma unroll
  for (int v = 0; v < 8; ++v) fgv[v] = fg[b * NQ + i0 + v + 8 * half];

  float rmax[8], rsum[8];
  #pragma unroll
  for (int v = 0; v < 8; ++v) rmax[v] = -FLT_MAX;

  #pragma unroll
  for (int jt = 0; jt < 6; ++jt) {
    int j = jt * 16 + l15;
    #pragma unroll
    for (int v = 0; v < 8; ++v) {
      float val = s[jt][v] * 0.125f;
      bool valid = (j < 77) ||
                   (j < 81 ? (fgv[v] != 0) : (j < CTX ? (fgv[v] == 0) : false));
      val = valid ? val : -FLT_MAX;
      s[jt][v] = val;
      rmax[v] = fmaxf(rmax[v], val);
    }
  }

  // ---- softmax: row reductions across the 16-lane half-wave ----
  #pragma unroll
  for (int v = 0; v < 8; ++v) {
    float m = rmax[v];
    m = fmaxf(m, __shfl_xor(m, 1));
    m = fmaxf(m, __shfl_xor(m, 2));
    m = fmaxf(m, __shfl_xor(m, 4));
    m = fmaxf(m, __shfl_xor(m, 8));
    rmax[v] = m;
    rsum[v] = 0.f;
  }
  #pragma unroll
  for (int jt = 0; jt < 6; ++jt) {
    #pragma unroll
    for (int v = 0; v < 8; ++v) {
      float p = __expf(s[jt][v] - rmax[v]);   // pad/masked cols -> 0
      s[jt][v] = p;
      rsum[v] += p;
    }
  }
  #pragma unroll
  for (int v = 0; v < 8; ++v) {
    float t = rsum[v];
    t += __shfl_xor(t, 1);
    t += __shfl_xor(t, 2);
    t += __shfl_xor(t, 4);
    t += __shfl_xor(t, 8);
    rsum[v] = 1.0f / t;
  }

  // ---- re-layout P (C-format) -> LDS row-major for A-format reads ----
  #pragma unroll
  for (int jt = 0; jt < 6; ++jt)
    #pragma unroll
    for (int v = 0; v < 8; ++v)
      plds[wv][(v + 8 * half) * 96 + jt * 16 + l15] = s[jt][v] * rsum[v];
  __syncthreads();

  // ---- out = P(16x96) * V(96x64); pad rows of V clamped (P==0 there) ----
  v8f o[4] = {};
  const float* pl = plds[wv];
  for (int kb = 0; kb < 96; kb += 4) {
    v2f a = *(const v2f*)(pl + l15 * 96 + kb + khalf);  // ds_load_b64
    int j0 = kb + khalf;
    int jc0 = j0     < CTX ? j0     : CTX - 1;
    int jc1 = j0 + 1 < CTX ? j0 + 1 : CTX - 1;
    const float* vp0 = vbuf + ((size_t)(b * CTX + jc0)) * INNER + h * DHEAD + l15;
    const float* vp1 = vbuf + ((size_t)(b * CTX + jc1)) * INNER + h * DHEAD + l15;
    #pragma unroll
    for (int nt = 0; nt < 4; ++nt) {
      v2f bf;
      bf.x = vp0[nt * 16];
      bf.y = vp1[nt * 16];
      o[nt] = wmma4(a, bf, o[nt]);
    }
  }

  #pragma unroll
  for (int nt = 0; nt < 4; ++nt)
    #pragma unroll
    for (int v = 0; v < 8; ++v)
      out[((size_t)(b * NQ + i0 + v + 8 * half)) * INNER + h * DHEAD + nt * 16 + l15] =
          o[nt][v];
}

// -------------------------------------------------------------------------
extern "C" void kernel_launch(void* const* d_in, const int* in_sizes, int n_in,
                              void* d_out, int out_size, void* d_ws, size_t ws_size,
                              hipStream_t stream) {
  (void)in_sizes; (void)n_in; (void)out_size; (void)ws_size;
  const float* x       = (const float*)d_in[0];
  const float* context = (const float*)d_in[1];
  const float* Wq      = (const float*)d_in[2];
  const float* Wk      = (const float*)d_in[3];
  const float* Wv      = (const float*)d_in[4];
  const float* Wo      = (const float*)d_in[5];
  const float* bo      = (const float*)d_in[6];
  const int*   mask    = (const int*)d_in[7];
  float* out = (float*)d_out;

  float* ws = (float*)d_ws;
  size_t qoff  = 0;
  size_t koff  = qoff + (size_t)BSZ * NQ * INNER;
  size_t voff  = koff + (size_t)BSZ * CTX * INNER;
  size_t aoff  = voff + (size_t)BSZ * CTX * INNER;
  size_t fgoff = aoff + (size_t)BSZ * NQ * INNER;
  float* qbuf = ws + qoff;
  float* kbuf = ws + koff;
  float* vbuf = ws + voff;
  float* abuf = ws + aoff;
  int*   fgbuf = (int*)(ws + fgoff);

  // 1. fg map from bicubic-resized mask
  hipLaunchKernelGGL(fg_kernel, dim3((BSZ * 4096 + 255) / 256), dim3(256), 0, stream,
                     mask, fgbuf);
  // 2. Q = x @ Wq            [32768,1280] x [1280,1280]
  hipLaunchKernelGGL(gemm_kernel, dim3(INNER / 256, (BSZ * NQ) / 128), dim3(256), 0, stream,
                     x, Wq, (const float*)nullptr, qbuf, BSZ * NQ, QD, INNER);
  // 3. K = context @ Wk      [680,1024] x [1024,1280]
  hipLaunchKernelGGL(gemm_kernel, dim3(INNER / 256, (BSZ * CTX + 127) / 128), dim3(256), 0, stream,
                     context, Wk, (const float*)nullptr, kbuf, BSZ * CTX, CDIM, INNER);
  // 4. V = context @ Wv
  hipLaunchKernelGGL(gemm_kernel, dim3(INNER / 256, (BSZ * CTX + 127) / 128), dim3(256), 0, stream,
                     context, Wv, (const float*)nullptr, vbuf, BSZ * CTX, CDIM, INNER);
  // 5. fused masked attention
  hipLaunchKernelGGL(attn_kernel, dim3(BSZ * HEADS * (NQ / 16) / 8), dim3(256), 0, stream,
                     qbuf, kbuf, vbuf, fgbuf, abuf);
  // 6. out = attn @ Wo + bo
  hipLaunchKernelGGL(gemm_kernel, dim3(QD / 256, (BSZ * NQ) / 128), dim3(256), 0, stream,
                     abuf, Wo, bo, out, BSZ * NQ, INNER, QD);
}